// TreatmentGNN_37692632990059
// MI455X (gfx1250) — compile-verified
//
#include <hip/hip_runtime.h>
#include <hip/hip_bf16.h>

typedef float v2f __attribute__((ext_vector_type(2)));
typedef float v8f __attribute__((ext_vector_type(8)));

// ---------------------------------------------------------------- norm setup
__global__ __launch_bounds__(256) void k_fill_deg(float* __restrict__ deg, int n) {
  int i = blockIdx.x * blockDim.x + threadIdx.x;
  if (i < n) deg[i] = 1.0f;                       // self-loop weight = 1
}

__global__ __launch_bounds__(256) void k_deg_scatter(const int* __restrict__ ei,
                                                     const float* __restrict__ w,
                                                     float* __restrict__ deg, int E) {
  int e = blockIdx.x * blockDim.x + threadIdx.x;
  if (e < E) atomicAdd(&deg[ei[E + e]], w[e]);    // dst = ei[E + e]
}

__global__ __launch_bounds__(256) void k_dinv(const float* __restrict__ deg,
                                              float* __restrict__ dinv, int n) {
  int i = blockIdx.x * blockDim.x + threadIdx.x;
  if (i < n) { float d = deg[i]; dinv[i] = d > 0.f ? rsqrtf(d) : 0.f; }
}

__global__ __launch_bounds__(256) void k_norm(const int* __restrict__ ei,
                                              const float* __restrict__ w,
                                              const float* __restrict__ dinv,
                                              float* __restrict__ norm, int E) {
  int e = blockIdx.x * blockDim.x + threadIdx.x;
  if (e < E) norm[e] = dinv[ei[e]] * w[e] * dinv[ei[E + e]];
}

// ---------------------------------------------------------------- WMMA GEMM
// H[N,fo] = act(X[N,fi]) @ W[fi,fo], act = relu if applyRelu.
// One wave per 16x64 output tile; K-loop in steps of 4 with V_WMMA_F32_16X16X4_F32.
// A frag (16x4 f32, 2 VGPR): lane l -> M = l&15, K = kk + 2*(l>>4) + j.
// B frag (4x16 f32, 2 VGPR): lane l -> N = l&15, K = kk + 2*(l>>4) + j (mirrors C/D pattern).
// C/D (16x16 f32, 8 VGPR):   elem r -> row = r + 8*(l>>4), col = l&15.
__global__ __launch_bounds__(256) void k_gemm_wmma(const float* __restrict__ X,
                                                   const float* __restrict__ Wm,
                                                   float* __restrict__ H,
                                                   int nRows, int fi, int fo, int applyRelu) {
  const int lane = threadIdx.x & 31;
  const int wave = threadIdx.x >> 5;
  const int wavesPerBlock = blockDim.x >> 5;
  const int tile = blockIdx.x * wavesPerBlock + wave;      // wave-uniform
  const int colTiles = fo >> 6;                            // 16x64 tiles per row band
  const int totalTiles = (nRows >> 4) * colTiles;
  if (tile >= totalTiles) return;                          // uniform per wave: EXEC stays full

  const int mBase = (tile / colTiles) << 4;
  const int nBase = (tile % colTiles) << 6;
  const int half = lane >> 4;
  const int lr = lane & 15;

  v8f acc0 = {}, acc1 = {}, acc2 = {}, acc3 = {};
  const float* __restrict__ xrow = X + (size_t)(mBase + lr) * fi;

  for (int kk = 0; kk < fi; kk += 4) {
    const int k0 = kk + 2 * half;
    float a0 = xrow[k0];
    float a1 = xrow[k0 + 1];
    if (applyRelu) { a0 = fmaxf(a0, 0.f); a1 = fmaxf(a1, 0.f); }
    v2f a = {a0, a1};

    const float* __restrict__ w0 = Wm + (size_t)k0 * fo + nBase + lr;
    const float* __restrict__ w1 = w0 + fo;
    v2f b0 = {w0[0],  w1[0]};
    v2f b1 = {w0[16], w1[16]};
    v2f b2 = {w0[32], w1[32]};
    v2f b3 = {w0[48], w1[48]};

    acc0 = __builtin_amdgcn_wmma_f32_16x16x4_f32(false, a, false, b0, (short)0, acc0, false, false);
    acc1 = __builtin_amdgcn_wmma_f32_16x16x4_f32(false, a, false, b1, (short)0, acc1, false, false);
    acc2 = __builtin_amdgcn_wmma_f32_16x16x4_f32(false, a, false, b2, (short)0, acc2, false, false);
    acc3 = __builtin_amdgcn_wmma_f32_16x16x4_f32(false, a, false, b3, (short)0, acc3, false, false);
  }

#pragma unroll
  for (int r = 0; r < 8; ++r) {
    const int row = mBase + r + 8 * half;
    float* __restrict__ o = H + (size_t)row * fo + nBase + lr;
    o[0]  = acc0[r];
    o[16] = acc1[r];
    o[32] = acc2[r];
    o[48] = acc3[r];
  }
}

// out[n,c] = bias[c] + dinv[n]^2 * h[n,c]   (bias + fused self-loop message)
__global__ __launch_bounds__(256) void k_layer_init(float* __restrict__ Out,
                                                    const float* __restrict__ H,
                                                    const float* __restrict__ bias,
                                                    const float* __restrict__ dinv,
                                                    int n, int fo) {
  long long i = (long long)blockIdx.x * blockDim.x + threadIdx.x;
  long long total = (long long)n * fo;
  if (i >= total) return;
  int node = (int)(i / fo);
  int c = (int)(i % fo);
  float di = dinv[node];
  Out[i] = bias[c] + di * di * H[i];
}

// One wave per edge: out[dst, c] += norm[e] * h[src, c], lanes cover channels.
__global__ __launch_bounds__(256) void k_edge_scatter(const float* __restrict__ H,
                                                      float* __restrict__ Out,
                                                      const int* __restrict__ ei,
                                                      const float* __restrict__ norm,
                                                      int E, int fo) {
  int e = blockIdx.x * (blockDim.x >> 5) + (threadIdx.x >> 5);
  int lane = threadIdx.x & 31;
  if (e >= E) return;
  int s = ei[e];
  int d = ei[E + e];
  float nv = norm[e];
  const float* __restrict__ hs = H + (size_t)s * fo;
  float* __restrict__ od = Out + (size_t)d * fo;
  for (int c = lane; c < fo; c += 32)
    atomicAdd(&od[c], nv * hs[c]);
}

// ---------------------------------------------------------------- host side
extern "C" void kernel_launch(void* const* d_in, const int* in_sizes, int n_in,
                              void* d_out, int out_size, void* d_ws, size_t ws_size,
                              hipStream_t stream) {
  const int* ei       = (const int*)d_in[0];       // [2, E]
  const float* ew     = (const float*)d_in[1];     // [E]
  const float* emb    = (const float*)d_in[2];     // [N, 128]
  const float* W[5]   = {(const float*)d_in[3], (const float*)d_in[5], (const float*)d_in[7],
                         (const float*)d_in[9], (const float*)d_in[11]};
  const float* B[5]   = {(const float*)d_in[4], (const float*)d_in[6], (const float*)d_in[8],
                         (const float*)d_in[10], (const float*)d_in[12]};
  const int FI[5] = {128, 128, 256, 256, 128};
  const int FO[5] = {128, 256, 256, 128, 128};

  const int E = in_sizes[1];
  const int N = in_sizes[2] / 128;

  // workspace layout (floats): dinv[N] | norm[E] | buf0[N*256] | buf1[N*256]
  float* dinv = (float*)d_ws;
  float* norm = dinv + N;
  float* buf0 = norm + E;                  // h (GEMM output) each layer; deg scratch first
  float* buf1 = buf0 + (size_t)N * 256;    // aggregation output / next layer input

  const int T = 256;
  // ---- GCN normalization ----
  float* deg = buf0;  // temporary, dead before first GEMM
  hipLaunchKernelGGL(k_fill_deg,    dim3((N + T - 1) / T), dim3(T), 0, stream, deg, N);
  hipLaunchKernelGGL(k_deg_scatter, dim3((E + T - 1) / T), dim3(T), 0, stream, ei, ew, deg, E);
  hipLaunchKernelGGL(k_dinv,        dim3((N + T - 1) / T), dim3(T), 0, stream, deg, dinv, N);
  hipLaunchKernelGGL(k_norm,        dim3((E + T - 1) / T), dim3(T), 0, stream, ei, ew, dinv, norm, E);

  // ---- 5 GCN layers ----
  const float* x = emb;
  for (int L = 0; L < 5; ++L) {
    const int fi = FI[L], fo = FO[L];
    float* h   = buf0;
    float* out = (L == 4) ? (float*)d_out : buf1;

    const int wavesPerBlock = T / 32;
    const int tiles = (N / 16) * (fo / 64);
    hipLaunchKernelGGL(k_gemm_wmma, dim3((tiles + wavesPerBlock - 1) / wavesPerBlock), dim3(T), 0,
                       stream, x, W[L], h, N, fi, fo, (L > 0) ? 1 : 0);

    long long tot = (long long)N * fo;
    hipLaunchKernelGGL(k_layer_init, dim3((unsigned)((tot + T - 1) / T)), dim3(T), 0, stream,
                       out, h, B[L], dinv, N, fo);

    hipLaunchKernelGGL(k_edge_scatter, dim3((E + wavesPerBlock - 1) / wavesPerBlock), dim3(T), 0,
                       stream, h, out, ei, norm, E, fo);

    x = out;  // ReLU folded into next layer's GEMM A-load; last layer has no ReLU
  }
}